// PSRoIPool_31791347925215
// MI455X (gfx1250) — compile-verified
//
#include <hip/hip_runtime.h>
#include <hip/hip_bf16.h>
#include <math.h>

// PSRoIPooling, direct-gather formulation (memory-bound; WMMA inapplicable
// because the source channel c = d*49 + ph*7 + pw varies per output bin, so
// there is no shared GEMM operand — dense-mask einsum would be ~800x more MACs).
//
// Layout facts (from reference):
//   features: [N=2, C=3969, H=64, W=64] f32
//   rois:     [R=256, 5] f32  (b, x1, y1, x2, y2) in image pixels
//   out:      [R, 81, 7, 7] f32, and within one ROI the flat output index i
//             equals the source channel index:  c = d*49 + ph*7 + pw = i.

#define PSR_N   2
#define PSR_C   3969
#define PSR_H   64
#define PSR_W   64
#define PSR_P   7
#define PSR_Q   7
#define PSR_R   256
#define PSR_SCALE (1.0f / 16.0f)
#define PSR_PER_ROI (PSR_C)          // 81*7*7 == 3969 outputs per ROI
#define PSR_YCHUNKS 4

// Address-space-qualified int pointers (builtin signature wants int*, per the
// round-1 diagnostic: "parameter of type '__device__ int *'").
typedef __attribute__((address_space(1))) int gas_int;
typedef __attribute__((address_space(3))) int las_int;

__global__ __launch_bounds__(256) void psroi_pool_kernel(
    const float* __restrict__ features,   // [N,C,H,W]
    const float* __restrict__ rois,       // [R,5]
    float* __restrict__ out)              // [R, 3969]
{
    const int r  = blockIdx.x;            // one ROI per blockIdx.x
    const int by = blockIdx.y;            // chunk of the 3969 outputs
    __shared__ float s_roi[8];

    // ---- Stage the 5-float ROI descriptor into LDS via the CDNA5 async
    // ---- global->LDS path (ASYNCcnt), one b32 per lane for lanes 0..4.
#if defined(__gfx1250__) && __has_builtin(__builtin_amdgcn_global_load_async_to_lds_b32)
    if (threadIdx.x < 5) {
        const float* g = rois + (size_t)r * 5 + threadIdx.x;
        __builtin_amdgcn_global_load_async_to_lds_b32(
            (gas_int*)(int*)(void*)(float*)g,
            (las_int*)(int*)(void*)&s_roi[threadIdx.x],
            /*imm offset*/ 0, /*cpol*/ 0);
    }
  #if __has_builtin(__builtin_amdgcn_s_wait_asynccnt)
    __builtin_amdgcn_s_wait_asynccnt(0);
  #else
    asm volatile("s_wait_asynccnt 0" ::: "memory");
  #endif
#else
    if (threadIdx.x < 5) {
        s_roi[threadIdx.x] = rois[(size_t)r * 5 + threadIdx.x];
    }
#endif
    __syncthreads();

    // ---- ROI geometry (reference semantics, f32 math, RNE rounding).
    const int   b       = (int)s_roi[0];
    const float start_w = rintf(s_roi[1]) * PSR_SCALE;
    const float start_h = rintf(s_roi[2]) * PSR_SCALE;
    const float end_w   = (rintf(s_roi[3]) + 1.0f) * PSR_SCALE;
    const float end_h   = (rintf(s_roi[4]) + 1.0f) * PSR_SCALE;
    const float roi_w   = fmaxf(end_w - start_w, 0.1f);
    const float roi_h   = fmaxf(end_h - start_h, 0.1f);
    const float bin_w   = roi_w * (1.0f / (float)PSR_Q);
    const float bin_h   = roi_h * (1.0f / (float)PSR_P);

    const size_t batch_base = (size_t)b * PSR_C * PSR_H * PSR_W;

    const int chunk = (PSR_PER_ROI + PSR_YCHUNKS - 1) / PSR_YCHUNKS;  // 993
    const int i_lo  = by * chunk;
    const int i_hi  = min(PSR_PER_ROI, i_lo + chunk);

    for (int i = i_lo + (int)threadIdx.x; i < i_hi; i += (int)blockDim.x) {
        const int rem = i % (PSR_P * PSR_Q);
        const int ph  = rem / PSR_Q;
        const int pw  = rem - ph * PSR_Q;
        // channel index == within-ROI output index (identity group gather)
        const int c = i;

        // Bin bounds, clipped to [0, H] / [0, W] as in the reference masks.
        float fhs = floorf((float)ph * bin_h + start_h);
        float fhe = ceilf(((float)ph + 1.0f) * bin_h + start_h);
        float fws = floorf((float)pw * bin_w + start_w);
        float fwe = ceilf(((float)pw + 1.0f) * bin_w + start_w);
        const int hs = (int)fminf(fmaxf(fhs, 0.0f), (float)PSR_H);
        const int he = (int)fminf(fmaxf(fhe, 0.0f), (float)PSR_H);
        const int ws = (int)fminf(fmaxf(fws, 0.0f), (float)PSR_W);
        const int we = (int)fminf(fmaxf(fwe, 0.0f), (float)PSR_W);

        const int hlen = he - hs;
        const int wlen = we - ws;

        float sum = 0.0f;
        if (hlen > 0 && wlen > 0) {
            const float* __restrict__ plane =
                features + batch_base + (size_t)c * (PSR_H * PSR_W);
            const float* __restrict__ row = plane + hs * PSR_W + ws;
            for (int h = 0; h < hlen; ++h) {
                if (h + 1 < hlen) {
                    // CDNA5 global_prefetch_b8 for the next bin row
                    __builtin_prefetch(row + PSR_W, 0, 1);
                }
                float rsum = 0.0f;
                #pragma unroll 4
                for (int w = 0; w < wlen; ++w) {
                    rsum += row[w];
                }
                sum += rsum;
                row += PSR_W;
            }
        }

        const float count = (float)(hlen * wlen);
        out[(size_t)r * PSR_PER_ROI + i] = (count > 0.0f) ? (sum / count) : 0.0f;
    }
}

extern "C" void kernel_launch(void* const* d_in, const int* in_sizes, int n_in,
                              void* d_out, int out_size, void* d_ws, size_t ws_size,
                              hipStream_t stream) {
    (void)in_sizes; (void)n_in; (void)d_ws; (void)ws_size; (void)out_size;
    const float* features = (const float*)d_in[0];   // [2,3969,64,64] f32
    const float* rois     = (const float*)d_in[1];   // [256,5] f32
    float*       out      = (float*)d_out;           // [256,81,7,7] f32

    dim3 grid(PSR_R, PSR_YCHUNKS, 1);   // 1024 blocks
    dim3 block(256, 1, 1);              // 8 wave32 per block
    psroi_pool_kernel<<<grid, block, 0, stream>>>(features, rois, out);
}